// LocalGrouper_12206297055630
// MI455X (gfx1250) — compile-verified
//
#include <hip/hip_runtime.h>
#include <math.h>

#define Bc   8
#define Nc   8192
#define CHc  64
#define Sc   2048
#define Kc   24
#define CF   67    // CH+3
#define CO   131   // CH+3+CH
#define CHUNK 512

typedef __attribute__((ext_vector_type(2))) float v2f;
typedef __attribute__((ext_vector_type(8))) float v8f;

// ---------------------------------------------------------------------------
// Kernel 1: farthest point sampling. 1 block per batch, 1024 threads.
// 2 barriers per serial step (critical path). Emits new_xyz (d_out) and
// gathered new_points (workspace); zero-inits per-batch f64 accumulators.
// ---------------------------------------------------------------------------
__global__ void fps_kernel(const float* __restrict__ xyz,
                           const float* __restrict__ points,
                           float* __restrict__ new_xyz,
                           float* __restrict__ new_points,
                           double* __restrict__ acc) {
    const int b    = blockIdx.x;
    const int tid  = threadIdx.x;
    const int lane = tid & 31;
    const int wave = tid >> 5;

    __shared__ int   sFar;
    __shared__ float redv[32];
    __shared__ int   redi[32];

    if (tid == 0) {
        sFar = 0;
        acc[b * 2 + 0] = 0.0;
        acc[b * 2 + 1] = 0.0;
    }

    // each thread owns 8 points: p = j*1024 + tid
    float px[8], py[8], pz[8], mind[8];
#pragma unroll
    for (int j = 0; j < 8; ++j) {
        const int p = j * 1024 + tid;
        const float* q = &xyz[((size_t)b * Nc + p) * 3];
        px[j] = q[0]; py[j] = q[1]; pz[j] = q[2];
        mind[j] = 1e10f;
    }
    __syncthreads();

    int far = 0;
    for (int s = 0; s < Sc; ++s) {
        // gather outputs for the current farthest point (off critical path)
        if (tid < CHc)
            new_points[((size_t)b * Sc + s) * CHc + tid] =
                points[((size_t)b * Nc + far) * CHc + tid];
        if (tid < 3)
            new_xyz[((size_t)b * Sc + s) * 3 + tid] =
                xyz[((size_t)b * Nc + far) * 3 + tid];

        // centroid: uniform-address loads, broadcast from L0
        const float* c = &xyz[((size_t)b * Nc + far) * 3];
        const float cx = c[0], cy = c[1], cz = c[2];

        float bestv = -1.0f;
        int   besti = 0;
#pragma unroll
        for (int j = 0; j < 8; ++j) {
            const float dx = px[j] - cx, dy = py[j] - cy, dz = pz[j] - cz;
            const float d = dx * dx + dy * dy + dz * dz;
            mind[j] = fminf(mind[j], d);
            if (mind[j] > bestv) { bestv = mind[j]; besti = j * 1024 + tid; }
        }
        // wave32 argmax (ties -> lower index)
#pragma unroll
        for (int off = 16; off > 0; off >>= 1) {
            const float ov = __shfl_xor(bestv, off);
            const int   oi = __shfl_xor(besti, off);
            if (ov > bestv || (ov == bestv && oi < besti)) { bestv = ov; besti = oi; }
        }
        if (lane == 0) { redv[wave] = bestv; redi[wave] = besti; }
        __syncthreads();
        if (wave == 0) {
            bestv = redv[lane]; besti = redi[lane];
#pragma unroll
            for (int off = 16; off > 0; off >>= 1) {
                const float ov = __shfl_xor(bestv, off);
                const int   oi = __shfl_xor(besti, off);
                if (ov > bestv || (ov == bestv && oi < besti)) { bestv = ov; besti = oi; }
            }
            if (lane == 0) sFar = besti;
        }
        __syncthreads();
        far = sFar;
    }
}

// ---------------------------------------------------------------------------
// Kernel 2: KNN via V_WMMA_F32_16X16X4_F32 with async-to-LDS double-buffered
// staging of dst xyz, + top-K selection + std partials.
// Grid = B * (S/16) blocks, 512 threads (16 waves). Block covers 16 src rows.
//   dist tile: D = (-2*src) . dst + |dst|^2 (via K slot 3) + |src|^2 (via C).
// ---------------------------------------------------------------------------
__global__ void knn_kernel(const float* __restrict__ xyz,
                           const float* __restrict__ points,
                           const float* __restrict__ new_xyz,
                           const float* __restrict__ new_points,
                           int* __restrict__ knn_idx,
                           double* __restrict__ acc) {
    const int blk  = blockIdx.x;
    const int b    = blk >> 7;            // S/16 = 128 tiles per batch
    const int s0   = (blk & 127) * 16;
    const int tid  = threadIdx.x;
    const int lane = tid & 31;
    const int wave = tid >> 5;
    const int mlane = lane & 15;

    __shared__ __align__(16) float sXYZ[2][CHUNK * 3];  // double-buffered dst coords
    __shared__ float  sSrc[48];           // 16 src points xyz
    __shared__ float  sNorm[16];          // |src|^2
    __shared__ float  sDist[16 * CHUNK];  // distance tile chunk
    __shared__ float  sTopD[16 * Kc];
    __shared__ int    sTopI[16 * Kc];
    __shared__ double sRed[512];
    __shared__ double sRed2[512];

    if (tid < 48) sSrc[tid] = new_xyz[((size_t)b * Sc + s0) * 3 + tid];
    __syncthreads();
    if (tid < 16) {
        const float x = sSrc[tid * 3], y = sSrc[tid * 3 + 1], z = sSrc[tid * 3 + 2];
        sNorm[tid] = x * x + y * y + z * z;
    }
    if (tid < 16 * Kc) { sTopD[tid] = 3.4e38f; sTopI[tid] = 0; }
    __syncthreads();

    // A fragment (shared by every tile of this block), branchless.
    const bool lo = (lane < 16);
    v2f afrag;
    afrag.x = -2.0f * (lo ? sSrc[mlane * 3 + 0] : sSrc[mlane * 3 + 2]);
    afrag.y = lo ? (-2.0f * sSrc[mlane * 3 + 1]) : 1.0f;
    const int rb = lo ? 0 : 8;
    v8f cfrag;
#pragma unroll
    for (int r = 0; r < 8; ++r) cfrag[r] = sNorm[rb + r];

    // async staging: 512 pts * 12B = 6144B = 384 x b128 (waves 0..11, all lanes)
    const uint32_t ldsBase0 = (uint32_t)(uintptr_t)&sXYZ[0][0];
    const uint32_t ldsBase1 = (uint32_t)(uintptr_t)&sXYZ[1][0];
#define STAGE(chunkIdx, bufIdx)                                                  \
    do {                                                                         \
        if (tid < (CHUNK * 12) / 16) {                                           \
            const unsigned long long gbase = (unsigned long long)(uintptr_t)     \
                (xyz + ((size_t)b * Nc + (chunkIdx) * CHUNK) * 3);               \
            const uint32_t voff  = (uint32_t)tid * 16u;                          \
            const uint32_t laddr = ((bufIdx) ? ldsBase1 : ldsBase0) + voff;      \
            asm volatile("global_load_async_to_lds_b128 %0, %1, %2"              \
                         :: "v"(laddr), "v"(voff), "s"(gbase) : "memory");       \
        }                                                                        \
    } while (0)

    STAGE(0, 0);

    for (int chunk = 0; chunk < Nc / CHUNK; ++chunk) {
        const int base = chunk * CHUNK;
        const int buf  = chunk & 1;

        asm volatile("s_wait_asynccnt 0x0" ::: "memory");
        __syncthreads();  // staged buf visible; sDist free for reuse

        if (chunk + 1 < Nc / CHUNK) STAGE(chunk + 1, buf ^ 1);

        // --- compute phase: 32 tiles of 16 dst cols; wave w does tiles w, w+16
        const float* dbuf = &sXYZ[buf][0];
        for (int tile = wave; tile < CHUNK / 16; tile += 16) {
            const float* q = &dbuf[(tile * 16 + mlane) * 3];
            const float qx = q[0], qy = q[1], qz = q[2];
            const float nrm = qx * qx + qy * qy + qz * qz;
            v2f bfrag;
            bfrag.x = lo ? qx : qz;
            bfrag.y = lo ? qy : nrm;

            v8f dmat = __builtin_amdgcn_wmma_f32_16x16x4_f32(
                false, afrag, false, bfrag, (short)0, cfrag, false, false);

            const int col = tile * 16 + mlane;
#pragma unroll
            for (int r = 0; r < 8; ++r) sDist[(rb + r) * CHUNK + col] = dmat[r];
        }
        __syncthreads();

        // --- selection phase: wave w owns src row w
        {
            float* topd_row = &sTopD[wave * Kc];
            int*   topi_row = &sTopI[wave * Kc];
            for (int i = lane; i < CHUNK; i += 32) {
                const float d  = sDist[wave * CHUNK + i];
                const int   gi = base + i;
                const float worst = topd_row[Kc - 1];
                unsigned long long m = __ballot(d < worst);
                while (m) {
                    const int j = __ffsll(m) - 1;
                    m &= m - 1;
                    const float dj = __shfl(d, j);
                    const int   ij = __shfl(gi, j);
                    const float cw = topd_row[Kc - 1];
                    if (dj < cw) {
                        if (lane == 0) {
                            int p = Kc - 1;
                            while (p > 0 && topd_row[p - 1] > dj) {
                                topd_row[p] = topd_row[p - 1];
                                topi_row[p] = topi_row[p - 1];
                                --p;
                            }
                            topd_row[p] = dj;
                            topi_row[p] = ij;
                        }
                    }
                }
            }
        }
    }
    __syncthreads();  // sTopI complete across all waves

    // write knn indices
    for (int e = tid; e < 16 * Kc; e += 512) {
        const int r = e / Kc, k = e % Kc;
        knn_idx[((size_t)b * Sc + s0 + r) * Kc + k] = sTopI[r * Kc + k];
    }

    // --- std partials: sum / sumsq of centered features for these 16 rows
    double lsum = 0.0, lsq = 0.0;
    const int TOT = 16 * Kc * CF;  // 25728
    for (int e = tid; e < TOT; e += 512) {
        const int ch = e % CF;
        const int rk = e / CF;
        const int k  = rk % Kc;
        const int r  = rk / Kc;
        const int pi = sTopI[r * Kc + k];
        float g, anc;
        if (ch < CHc) {
            g   = points[((size_t)b * Nc + pi) * CHc + ch];
            anc = new_points[((size_t)b * Sc + s0 + r) * CHc + ch];
        } else {
            const int c3 = ch - CHc;
            g   = xyz[((size_t)b * Nc + pi) * 3 + c3];
            anc = sSrc[r * 3 + c3];
        }
        const float cen = g - anc;
        lsum += (double)cen;
        lsq  += (double)cen * (double)cen;
    }
    sRed[tid] = lsum; sRed2[tid] = lsq;
    __syncthreads();
    for (int off = 256; off > 0; off >>= 1) {
        if (tid < off) { sRed[tid] += sRed[tid + off]; sRed2[tid] += sRed2[tid + off]; }
        __syncthreads();
    }
    if (tid == 0) {
        atomicAdd(&acc[b * 2 + 0], sRed[0]);
        atomicAdd(&acc[b * 2 + 1], sRed2[0]);
    }
#undef STAGE
}

// ---------------------------------------------------------------------------
// Kernel 3: per-batch 1/(std+eps), ddof=1.
// ---------------------------------------------------------------------------
__global__ void std_kernel(const double* __restrict__ acc, float* __restrict__ stdinv) {
    const int b = threadIdx.x;
    if (b < Bc) {
        const double m   = (double)Sc * Kc * CF;
        const double sum = acc[b * 2 + 0];
        const double sq  = acc[b * 2 + 1];
        double var = (sq - sum * sum / m) / (m - 1.0);
        if (var < 0.0) var = 0.0;
        stdinv[b] = (float)(1.0 / (sqrt(var) + 1e-5));
    }
}

// ---------------------------------------------------------------------------
// Kernel 4: gather + normalize + affine + anchor concat -> new_feats.
// ---------------------------------------------------------------------------
__global__ void finalize_kernel(const float* __restrict__ xyz,
                                const float* __restrict__ points,
                                const float* __restrict__ alpha,
                                const float* __restrict__ beta,
                                const float* __restrict__ new_points,
                                const int* __restrict__ knn_idx,
                                const float* __restrict__ stdinv,
                                const float* __restrict__ new_xyz,
                                float* __restrict__ out_feats) {
    const long long TOT = (long long)Bc * Sc * Kc * CO;
    const long long stride = (long long)gridDim.x * blockDim.x;
    for (long long t = (long long)blockIdx.x * blockDim.x + threadIdx.x; t < TOT; t += stride) {
        const int ch = (int)(t % CO);
        long long r = t / CO;
        const int k = (int)(r % Kc); r /= Kc;
        const int s = (int)(r % Sc); r /= Sc;
        const int b = (int)r;
        float val;
        if (ch < CF) {
            const int pi = knn_idx[((size_t)b * Sc + s) * Kc + k];
            float g, anc;
            if (ch < CHc) {
                g   = points[((size_t)b * Nc + pi) * CHc + ch];
                anc = new_points[((size_t)b * Sc + s) * CHc + ch];
            } else {
                g   = xyz[((size_t)b * Nc + pi) * 3 + (ch - CHc)];
                anc = new_xyz[((size_t)b * Sc + s) * 3 + (ch - CHc)];
            }
            val = (g - anc) * stdinv[b] * alpha[ch] + beta[ch];
        } else {
            val = new_points[((size_t)b * Sc + s) * CHc + (ch - CF)];
        }
        out_feats[t] = val;
    }
}

// ---------------------------------------------------------------------------
extern "C" void kernel_launch(void* const* d_in, const int* in_sizes, int n_in,
                              void* d_out, int out_size, void* d_ws, size_t ws_size,
                              hipStream_t stream) {
    const float* xyz    = (const float*)d_in[0];
    const float* points = (const float*)d_in[1];
    const float* alpha  = (const float*)d_in[2];
    const float* beta   = (const float*)d_in[3];

    float* out       = (float*)d_out;
    float* new_xyz   = out;                          // [B,S,3]
    float* new_feats = out + (size_t)Bc * Sc * 3;    // [B,S,K,131]

    char*   ws         = (char*)d_ws;
    float*  new_points = (float*)(ws);               // B*S*CH floats = 4 MiB
    int*    knn_idx    = (int*)(ws + 4194304);       // B*S*K ints   = 1.5 MiB
    double* acc        = (double*)(ws + 5767168);    // B*2 doubles
    float*  stdinv     = (float*)(ws + 5767296);     // B floats

    fps_kernel<<<Bc, 1024, 0, stream>>>(xyz, points, new_xyz, new_points, acc);
    knn_kernel<<<Bc * (Sc / 16), 512, 0, stream>>>(xyz, points, new_xyz, new_points,
                                                   knn_idx, acc);
    std_kernel<<<1, 32, 0, stream>>>(acc, stdinv);
    finalize_kernel<<<4096, 256, 0, stream>>>(xyz, points, alpha, beta, new_points,
                                              knn_idx, stdinv, new_xyz, new_feats);
}